// BaseCrossAttention_57028575756703
// MI455X (gfx1250) — compile-verified
//
#include <hip/hip_runtime.h>
#include <hip/hip_bf16.h>

// BaseCrossAttention fused implementation for gfx1250 (MI455X).
// All GEMMs on V_WMMA_F32_16X16X4_F32 (fp32 matrix pipe, wave32).
// Memory-bound (~40 GFLOP vs ~134 MB HBM): fuse per 32-row T-tile, reuse
// weight fragments across row-tiles, store K transposed for coalescing.

typedef float v2f __attribute__((ext_vector_type(2)));
typedef float v8f __attribute__((ext_vector_type(8)));

#define LN_EPS 1e-5f

__device__ __forceinline__ v8f wmma4(v2f a, v2f b, v8f c) {
  // (neg_a, A, neg_b, B, c_mod, C, reuse_a, reuse_b)
  return __builtin_amdgcn_wmma_f32_16x16x4_f32(false, a, false, b, (short)0, c,
                                               false, false);
}

__device__ __forceinline__ float sigmoidf_(float x) {
  return 1.0f / (1.0f + __expf(-x));
}

// ---------------------------------------------------------------------------
// Kernel A: per batch, LN(xf); K = ln@Wk^T+bk stored TRANSPOSED per head as
// kbufT[b][h][ch(64)][key(80)]; V = (ln@Wv^T+bv)*tc stored [b][key(80)][512].
// Keys padded 77->80 (pad LN rows are zero). One weight fragment feeds 5
// row-tile accumulators (5x weight-traffic reduction).
// ---------------------------------------------------------------------------
__global__ void kv_kernel(const float* __restrict__ xf,
                          const float* __restrict__ tnw,
                          const float* __restrict__ tnb,
                          const float* __restrict__ Wk,
                          const float* __restrict__ bk,
                          const float* __restrict__ Wv,
                          const float* __restrict__ bv,
                          const int*   __restrict__ cond_type,
                          float* __restrict__ kbufT,
                          float* __restrict__ vbuf) {
  extern __shared__ float sm[];              // [80][256] LN(xf)
  const int b    = blockIdx.x;
  const int tid  = threadIdx.x;
  const int lane = tid & 31;
  const int l15  = lane & 15;
  const int half = (lane >> 4) & 1;
  const int hw   = tid >> 4;                 // half-wave id 0..15
  const float tc = ((cond_type[b] % 10) > 0) ? 1.0f : 0.0f;

  // LayerNorm of xf rows (one row per half-wave, 5 sweeps for 80 rows)
  for (int it = 0; it < 5; ++it) {
    int row = it * 16 + hw;
    float* dst = sm + row * 256;
    if (row < 77) {
      const float* src = xf + ((size_t)b * 77 + row) * 256;
      float s = 0.f, ss = 0.f;
      float vals[16];
      #pragma unroll
      for (int j = 0; j < 16; ++j) {
        float v = src[l15 + 16 * j];
        vals[j] = v; s += v; ss += v * v;
      }
      #pragma unroll
      for (int m = 1; m < 16; m <<= 1) {
        s  += __shfl_xor(s,  m, 16);
        ss += __shfl_xor(ss, m, 16);
      }
      float mu  = s * (1.0f / 256.0f);
      float var = ss * (1.0f / 256.0f) - mu * mu;
      float inv = rsqrtf(var + LN_EPS);
      #pragma unroll
      for (int j = 0; j < 16; ++j) {
        int c = l15 + 16 * j;
        dst[c] = (vals[j] - mu) * inv * tnw[c] + tnb[c];
      }
    } else {
      #pragma unroll
      for (int j = 0; j < 16; ++j) dst[l15 + 16 * j] = 0.f;
    }
  }
  __syncthreads();

  const int wave = tid >> 5;                 // 0..7
  // 64 column-tile tasks: t<32 -> K columns, else V columns.
  // Each task computes all 5 row-tiles (80 keys) with one B stream.
  for (int t = wave; t < 64; t += 8) {
    const int isv  = (t >= 32) ? 1 : 0;
    const int nt   = t & 31;
    const int ncol = nt * 16 + l15;          // output channel 0..511
    const float* W    = isv ? Wv : Wk;
    const float* bias = isv ? bv : bk;
    const float2* Bp  = (const float2*)(W + (size_t)ncol * 256);
    v8f acc[5] = {};
    for (int k0 = 0; k0 < 256; k0 += 4) {
      int ka = k0 + 2 * half;
      float2 bw = Bp[ka >> 1];
      v2f bb; bb.x = bw.x; bb.y = bw.y;
      #pragma unroll
      for (int rt = 0; rt < 5; ++rt) {
        const float* Arow = sm + (rt * 16 + l15) * 256;
        v2f a; a.x = Arow[ka]; a.y = Arow[ka + 1];
        acc[rt] = wmma4(a, bb, acc[rt]);
      }
    }
    const float bias_n = bias[ncol];
    if (isv) {
      #pragma unroll
      for (int rt = 0; rt < 5; ++rt)
        #pragma unroll
        for (int r = 0; r < 8; ++r) {
          int key = rt * 16 + r + 8 * half;
          vbuf[((size_t)b * 80 + key) * 512 + ncol] = (acc[rt][r] + bias_n) * tc;
        }
    } else {
      const int h = ncol >> 6, ch = ncol & 63;
      float* kT = kbufT + (((size_t)b * 8 + h) * 64 + ch) * 80;
      #pragma unroll
      for (int rt = 0; rt < 5; ++rt)
        #pragma unroll
        for (int r = 0; r < 8; ++r) {
          int key = rt * 16 + r + 8 * half;
          kT[key] = acc[rt][r] + bias_n;
        }
    }
  }
}

// ---------------------------------------------------------------------------
// Kernel B: e = silu(emb) @ emb_w^T + emb_b -> scale/shift halves.
// Single block, 32 batch rows (2 row-tiles), one B stream feeds both.
// ---------------------------------------------------------------------------
__global__ void emb_kernel(const float* __restrict__ emb,
                           const float* __restrict__ emb_w,
                           const float* __restrict__ emb_b,
                           float* __restrict__ escale,
                           float* __restrict__ eshift) {
  extern __shared__ float sm[];              // [32][2048] silu(emb)
  const int tid = threadIdx.x;
  for (int i = tid; i < 32 * 2048; i += 256) {
    float v = emb[i];
    sm[i] = v * sigmoidf_(v);
  }
  __syncthreads();
  const int lane = tid & 31, l15 = lane & 15, half = (lane >> 4) & 1;
  const int wave = tid >> 5;
  for (int nt = wave; nt < 64; nt += 8) {
    const int ncol = nt * 16 + l15;
    const float2* Bp = (const float2*)(emb_w + (size_t)ncol * 2048);
    v8f acc0 = {}, acc1 = {};
    #pragma unroll 4
    for (int k0 = 0; k0 < 2048; k0 += 4) {
      int ka = k0 + 2 * half;
      float2 bw = Bp[ka >> 1];
      v2f bb; bb.x = bw.x; bb.y = bw.y;
      v2f a0; a0.x = sm[l15 * 2048 + ka];        a0.y = sm[l15 * 2048 + ka + 1];
      v2f a1; a1.x = sm[(16 + l15) * 2048 + ka]; a1.y = sm[(16 + l15) * 2048 + ka + 1];
      acc0 = wmma4(a0, bb, acc0);
      acc1 = wmma4(a1, bb, acc1);
    }
    const float bias_n = emb_b[ncol];
    #pragma unroll
    for (int r = 0; r < 8; ++r) {
      int row = r + 8 * half;
      float v0 = acc0[r] + bias_n;
      float v1 = acc1[r] + bias_n;
      if (ncol < 512) {
        escale[(size_t)row * 512 + ncol] = v0;
        escale[(size_t)(row + 16) * 512 + ncol] = v1;
      } else {
        eshift[(size_t)row * 512 + (ncol - 512)] = v0;
        eshift[(size_t)(row + 16) * 512 + (ncol - 512)] = v1;
      }
    }
  }
}

// ---------------------------------------------------------------------------
// Kernel C: fused per (batch, 32-row T-tile):
//   LN(x) -> q-proj -> per-head QK^T -> softmax -> AV -> LN(y) -> styliz ->
//   silu -> out-proj -> residual add.
// The (1-tc)*-1e6 attention bias is constant along the softmax axis and
// cancels exactly; gating flows through V (applied in kv_kernel).
// Weight/K/V fragments are reused across both 16-row tiles (2 accumulators).
// ---------------------------------------------------------------------------
__global__ void attn_kernel(const float* __restrict__ x,
                            const float* __restrict__ nw,
                            const float* __restrict__ nb,
                            const float* __restrict__ Wq,
                            const float* __restrict__ bq,
                            const float* __restrict__ snw,
                            const float* __restrict__ snb,
                            const float* __restrict__ Wout,
                            const float* __restrict__ bout,
                            const float* __restrict__ kbufT,
                            const float* __restrict__ vbuf,
                            const float* __restrict__ escale,
                            const float* __restrict__ eshift,
                            float* __restrict__ out) {
  extern __shared__ float sm[];
  float* smA = sm;              // 32*512: lnx, later y
  float* smB = sm + 16384;      // 32*512: q, later silu(h)
  float* smS = sm + 32768;      // 8*32*80: attention scores/weights
  const int tid  = threadIdx.x;
  const int lane = tid & 31, l15 = lane & 15, half = (lane >> 4) & 1;
  const int wave = tid >> 5;    // 0..7 (== head in attention stages)
  const int hw   = tid >> 4;    // 0..15 (row id for LN stages)
  const int b    = blockIdx.y;
  const int t0   = blockIdx.x * 32;

  // Stage 1: LN(x tile, 32 rows) -> smA
  for (int it = 0; it < 2; ++it) {
    const int row = it * 16 + hw;
    const float* src = x + (((size_t)b * 1024) + t0 + row) * 512;
    float s = 0.f, ss = 0.f;
    float vals[32];
    #pragma unroll
    for (int j = 0; j < 32; ++j) {
      float v = src[l15 + 16 * j];
      vals[j] = v; s += v; ss += v * v;
    }
    #pragma unroll
    for (int m = 1; m < 16; m <<= 1) {
      s += __shfl_xor(s, m, 16); ss += __shfl_xor(ss, m, 16);
    }
    float mu  = s * (1.0f / 512.0f);
    float var = ss * (1.0f / 512.0f) - mu * mu;
    float inv = rsqrtf(var + LN_EPS);
    #pragma unroll
    for (int j = 0; j < 32; ++j) {
      int c = l15 + 16 * j;
      smA[row * 512 + c] = (vals[j] - mu) * inv * nw[c] + nb[c];
    }
  }
  __syncthreads();

  // Stage 2: q = lnx @ Wq^T + bq -> smB (one B stream feeds 2 row-tiles)
  for (int nt = wave; nt < 32; nt += 8) {
    const int ncol = nt * 16 + l15;
    const float2* Bp = (const float2*)(Wq + (size_t)ncol * 512);
    v8f acc0 = {}, acc1 = {};
    #pragma unroll 4
    for (int k0 = 0; k0 < 512; k0 += 4) {
      int ka = k0 + 2 * half;
      float2 bw = Bp[ka >> 1];
      v2f bb; bb.x = bw.x; bb.y = bw.y;
      v2f a0; a0.x = smA[l15 * 512 + ka];        a0.y = smA[l15 * 512 + ka + 1];
      v2f a1; a1.x = smA[(16 + l15) * 512 + ka]; a1.y = smA[(16 + l15) * 512 + ka + 1];
      acc0 = wmma4(a0, bb, acc0);
      acc1 = wmma4(a1, bb, acc1);
    }
    const float bias_n = bq[ncol];
    #pragma unroll
    for (int r = 0; r < 8; ++r) {
      smB[(r + 8 * half) * 512 + ncol]      = acc0[r] + bias_n;
      smB[(16 + r + 8 * half) * 512 + ncol] = acc1[r] + bias_n;
    }
  }
  __syncthreads();

  // Stage 3: scores[h][m][key] = q_h @ k_h^T (wave == head), keys padded to
  // 80. K is stored transposed [ch][key], so lanes load contiguous keys.
  {
    const int h = wave;
    const float* kT = kbufT + ((size_t)b * 8 + h) * 64 * 80;
    for (int nt = 0; nt < 5; ++nt) {
      const int nkey = nt * 16 + l15;
      v8f acc0 = {}, acc1 = {};
      #pragma unroll
      for (int k0 = 0; k0 < 64; k0 += 4) {
        int ka = k0 + 2 * half;
        v2f bb; bb.x = kT[(size_t)ka * 80 + nkey];
                bb.y = kT[(size_t)(ka + 1) * 80 + nkey];
        v2f a0; a0.x = smB[l15 * 512 + h * 64 + ka];
                a0.y = smB[l15 * 512 + h * 64 + ka + 1];
        v2f a1; a1.x = smB[(16 + l15) * 512 + h * 64 + ka];
                a1.y = smB[(16 + l15) * 512 + h * 64 + ka + 1];
        acc0 = wmma4(a0, bb, acc0);
        acc1 = wmma4(a1, bb, acc1);
      }
      #pragma unroll
      for (int r = 0; r < 8; ++r) {
        smS[h * 2560 + (r + 8 * half) * 80 + nkey]      = acc0[r];
        smS[h * 2560 + (16 + r + 8 * half) * 80 + nkey] = acc1[r];
      }
    }
  }
  __syncthreads();

  // Stage 4: softmax over 77 keys (one (h,row) per thread), zero pad columns
  {
    const int h = tid >> 5, m = tid & 31;   // 8 heads x 32 rows = 256 threads
    float* srow = smS + h * 2560 + m * 80;
    float mx = -1e30f;
    for (int n = 0; n < 77; ++n) mx = fmaxf(mx, srow[n]);
    float ssum = 0.f;
    for (int n = 0; n < 77; ++n) { float e = __expf(srow[n] - mx); srow[n] = e; ssum += e; }
    float rinv = 1.0f / ssum;
    for (int n = 0; n < 77; ++n) srow[n] *= rinv;
    srow[77] = 0.f; srow[78] = 0.f; srow[79] = 0.f;
  }
  __syncthreads();

  // Stage 5: y_h = attn_h @ v_h (wave == head, 4 ch tiles x 2 row-tiles)
  {
    const int h = wave;
    const float* vb = vbuf + (size_t)b * 80 * 512 + h * 64;
    for (int nt = 0; nt < 4; ++nt) {
      const int ncol = nt * 16 + l15;
      v8f acc0 = {}, acc1 = {};
      #pragma unroll
      for (int k0 = 0; k0 < 80; k0 += 4) {
        int ka = k0 + 2 * half;
        v2f bb; bb.x = vb[(size_t)ka * 512 + ncol];
                bb.y = vb[(size_t)(ka + 1) * 512 + ncol];
        v2f a0; a0.x = smS[h * 2560 + l15 * 80 + ka];
                a0.y = smS[h * 2560 + l15 * 80 + ka + 1];
        v2f a1; a1.x = smS[h * 2560 + (16 + l15) * 80 + ka];
                a1.y = smS[h * 2560 + (16 + l15) * 80 + ka + 1];
        acc0 = wmma4(a0, bb, acc0);
        acc1 = wmma4(a1, bb, acc1);
      }
      #pragma unroll
      for (int r = 0; r < 8; ++r) {
        smA[(r + 8 * half) * 512 + h * 64 + ncol]      = acc0[r];
        smA[(16 + r + 8 * half) * 512 + h * 64 + ncol] = acc1[r];
      }
    }
  }
  __syncthreads();

  // Stage 6: LN(y) * (1+scale) + shift, then silu -> smB
  for (int it = 0; it < 2; ++it) {
    const int row = it * 16 + hw;
    float s = 0.f, ss = 0.f;
    float vals[32];
    #pragma unroll
    for (int j = 0; j < 32; ++j) {
      float v = smA[row * 512 + l15 + 16 * j];
      vals[j] = v; s += v; ss += v * v;
    }
    #pragma unroll
    for (int m = 1; m < 16; m <<= 1) {
      s += __shfl_xor(s, m, 16); ss += __shfl_xor(ss, m, 16);
    }
    float mu  = s * (1.0f / 512.0f);
    float var = ss * (1.0f / 512.0f) - mu * mu;
    float inv = rsqrtf(var + LN_EPS);
    #pragma unroll
    for (int j = 0; j < 32; ++j) {
      int c = l15 + 16 * j;
      float lny = (vals[j] - mu) * inv * snw[c] + snb[c];
      float hv  = lny * (1.0f + escale[(size_t)b * 512 + c]) +
                  eshift[(size_t)b * 512 + c];
      smB[row * 512 + c] = hv * sigmoidf_(hv);
    }
  }
  __syncthreads();

  // Stage 7: out = x + silu(h) @ Wout^T + bout (one B stream, 2 row-tiles)
  for (int nt = wave; nt < 32; nt += 8) {
    const int ncol = nt * 16 + l15;
    const float2* Bp = (const float2*)(Wout + (size_t)ncol * 512);
    v8f acc0 = {}, acc1 = {};
    #pragma unroll 4
    for (int k0 = 0; k0 < 512; k0 += 4) {
      int ka = k0 + 2 * half;
      float2 bw = Bp[ka >> 1];
      v2f bb; bb.x = bw.x; bb.y = bw.y;
      v2f a0; a0.x = smB[l15 * 512 + ka];        a0.y = smB[l15 * 512 + ka + 1];
      v2f a1; a1.x = smB[(16 + l15) * 512 + ka]; a1.y = smB[(16 + l15) * 512 + ka + 1];
      acc0 = wmma4(a0, bb, acc0);
      acc1 = wmma4(a1, bb, acc1);
    }
    const float bias_n = bout[ncol];
    #pragma unroll
    for (int r = 0; r < 8; ++r) {
      size_t i0 = (((size_t)b * 1024) + t0 + r + 8 * half) * 512 + ncol;
      size_t i1 = (((size_t)b * 1024) + t0 + 16 + r + 8 * half) * 512 + ncol;
      out[i0] = x[i0] + acc0[r] + bias_n;
      out[i1] = x[i1] + acc1[r] + bias_n;
    }
  }
}

// ---------------------------------------------------------------------------
extern "C" void kernel_launch(void* const* d_in, const int* in_sizes, int n_in,
                              void* d_out, int out_size, void* d_ws, size_t ws_size,
                              hipStream_t stream) {
  const float* x        = (const float*)d_in[0];
  const float* xf       = (const float*)d_in[1];
  const float* emb      = (const float*)d_in[2];
  // d_in[3] = src_mask (unused by reference)
  const int*   cond     = (const int*)  d_in[4];
  const float* norm_w   = (const float*)d_in[5];
  const float* norm_b   = (const float*)d_in[6];
  const float* tnorm_w  = (const float*)d_in[7];
  const float* tnorm_b  = (const float*)d_in[8];
  const float* Wq       = (const float*)d_in[9];
  const float* bq       = (const float*)d_in[10];
  const float* Wk       = (const float*)d_in[11];
  const float* bk       = (const float*)d_in[12];
  const float* Wv       = (const float*)d_in[13];
  const float* bv       = (const float*)d_in[14];
  const float* emb_w    = (const float*)d_in[15];
  const float* emb_b    = (const float*)d_in[16];
  const float* snorm_w  = (const float*)d_in[17];
  const float* snorm_b  = (const float*)d_in[18];
  const float* Wout     = (const float*)d_in[19];
  const float* bout     = (const float*)d_in[20];
  float* outp = (float*)d_out;

  // Workspace layout (floats)
  float* ws     = (float*)d_ws;
  float* kbufT  = ws;                               // 32*8*64*80
  float* vbuf   = kbufT + (size_t)32 * 8 * 64 * 80; // 32*80*512
  float* escale = vbuf + (size_t)32 * 80 * 512;     // 32*512
  float* eshift = escale + (size_t)32 * 512;        // 32*512

  kv_kernel<<<32, 256, 80 * 256 * 4, stream>>>(
      xf, tnorm_w, tnorm_b, Wk, bk, Wv, bv, cond, kbufT, vbuf);

  emb_kernel<<<1, 256, 32 * 2048 * 4, stream>>>(
      emb, emb_w, emb_b, escale, eshift);

  dim3 grid(32, 32);  // 32 T-tiles x 32 batches
  size_t lds = (size_t)(32 * 512 * 2 + 8 * 32 * 80) * 4;  // ~208 KB
  attn_kernel<<<grid, 256, lds, stream>>>(
      x, norm_w, norm_b, Wq, bq, snorm_w, snorm_b, Wout, bout,
      kbufT, vbuf, escale, eshift, outp);
}